// GraphSAGE_87720412053738
// MI455X (gfx1250) — compile-verified
//
#include <hip/hip_runtime.h>
#include <hip/hip_bf16.h>
#include <math.h>

typedef __attribute__((ext_vector_type(2))) float v2f;
typedef __attribute__((ext_vector_type(8))) float v8f;

#define DIN 128
#define LDA (DIN + 4)   // padded LDS row stride in dwords (bank-conflict-free)

// ---------------------------------------------------------------------------
// deg[i] = number of edges with dst == i (as float)
__global__ void degree_kernel(const int* __restrict__ dst, float* __restrict__ deg,
                              int n_edges) {
    int e = blockIdx.x * blockDim.x + threadIdx.x;
    if (e < n_edges) atomicAdd(&deg[dst[e]], 1.0f);
}

// ---------------------------------------------------------------------------
// agg[dst[e], :] += feat[src[e], :]   (one wave per edge, 4 floats per lane)
__global__ void scatter_add_kernel(const float* __restrict__ feat,
                                   const int* __restrict__ src,
                                   const int* __restrict__ dst,
                                   float* __restrict__ agg, int n_edges) {
    long long t = (long long)blockIdx.x * blockDim.x + threadIdx.x;
    // edge index is wave-uniform: force into SGPR so src/dst become s_loads
    // and the row gather uses SGPR base + lane offset (saddr form).
    int e    = __builtin_amdgcn_readfirstlane((int)(t >> 5));
    int lane = threadIdx.x & 31;
    if (e >= n_edges) return;                 // wave-uniform exit
    int s = __builtin_amdgcn_readfirstlane(src[e]);
    int d = __builtin_amdgcn_readfirstlane(dst[e]);
    const float4 v = *(const float4*)(feat + (size_t)s * DIN + lane * 4);
    float* ap = agg + (size_t)d * DIN + lane * 4;
    atomicAdd(ap + 0, v.x);
    atomicAdd(ap + 1, v.y);
    atomicAdd(ap + 2, v.z);
    atomicAdd(ap + 3, v.w);
}

// ---------------------------------------------------------------------------
// out[r,:] = act( (agg[r,:]/max(deg[r],1)) @ Wl + xin[r,:] @ Wr + b )
// CT = output col tiles (Dout = 16*CT), blockDim = 32*CT, 16 rows per block.
template <int CT, int RELU>
__global__ void sage_gemm_kernel(const float* __restrict__ agg,
                                 const float* __restrict__ xin,
                                 const float* __restrict__ deg,
                                 const float* __restrict__ Wl,
                                 const float* __restrict__ Wr,
                                 const float* __restrict__ bias,
                                 float* __restrict__ out,
                                 int n_rows) {
    constexpr int DOUT = 16 * CT;
    __shared__ float As[16 * LDA];
    __shared__ float Xs[16 * LDA];
    __shared__ float scal[16];

    const int tid = threadIdx.x;
    const int r0  = blockIdx.x * 16;

    if (tid < 16) {
        int r   = r0 + tid;
        float d = (r < n_rows) ? deg[r] : 1.0f;
        scal[tid] = 1.0f / fmaxf(d, 1.0f);
    }
    __syncthreads();

    // cooperative staging: 16 rows x 128 floats of agg (scaled) and xin
    for (int i = tid; i < 16 * (DIN / 4); i += 32 * CT) {
        int r  = i >> 5;           // i / 32
        int c4 = (i & 31) * 4;     // dword column
        int gr = r0 + r;
        float4 a4 = make_float4(0.f, 0.f, 0.f, 0.f);
        float4 x4 = make_float4(0.f, 0.f, 0.f, 0.f);
        if (gr < n_rows) {
            a4 = *(const float4*)(agg + (size_t)gr * DIN + c4);
            x4 = *(const float4*)(xin + (size_t)gr * DIN + c4);
        }
        float s = scal[r];
        float* ap = &As[r * LDA + c4];
        ap[0] = a4.x * s; ap[1] = a4.y * s; ap[2] = a4.z * s; ap[3] = a4.w * s;
        float* xp = &Xs[r * LDA + c4];
        xp[0] = x4.x; xp[1] = x4.y; xp[2] = x4.z; xp[3] = x4.w;
    }
    __syncthreads();

    const int wave  = tid >> 5;          // column tile owned by this wave
    const int lane  = tid & 31;
    const int row16 = lane & 15;
    const int khalf = (lane >> 4) * 2;   // lane half selects K pair {0,1}/{2,3}
    const int colg  = wave * 16 + row16; // this lane's B column

    // C init = bias broadcast down the column
    float bv = bias[colg];
    v8f c;
#pragma unroll
    for (int j = 0; j < 8; ++j) c[j] = bv;

#pragma unroll 4
    for (int kt = 0; kt < DIN / 4; ++kt) {
        int kb = kt * 4 + khalf;
        // A fragments (16x4 f32): lane half picks K pair, 8B-aligned LDS read
        v2f a = *(const v2f*)&As[row16 * LDA + kb];
        v2f x = *(const v2f*)&Xs[row16 * LDA + kb];
        // B fragments (4x16 f32): lane = column, lane half picks K pair
        v2f bl = { Wl[(size_t)kb * DOUT + colg], Wl[(size_t)(kb + 1) * DOUT + colg] };
        v2f br = { Wr[(size_t)kb * DOUT + colg], Wr[(size_t)(kb + 1) * DOUT + colg] };
        c = __builtin_amdgcn_wmma_f32_16x16x4_f32(false, a, false, bl, (short)0, c,
                                                  false, false);
        c = __builtin_amdgcn_wmma_f32_16x16x4_f32(false, x, false, br, (short)0, c,
                                                  false, false);
    }

    if (RELU) {
#pragma unroll
        for (int j = 0; j < 8; ++j) c[j] = fmaxf(c[j], 0.0f);
    }

    // C/D layout: VGPR j holds M = j + 8*(lane>=16), N = lane&15
    const int mbase = (lane >> 4) * 8;
    float* op = out + (size_t)(r0 + mbase) * DOUT + colg;
    if (r0 + 16 <= n_rows) {
        // fast path: whole row-block in range -> straight-line stores
#pragma unroll
        for (int j = 0; j < 8; ++j) op[(size_t)j * DOUT] = c[j];
    } else {
#pragma unroll
        for (int j = 0; j < 8; ++j)
            if (r0 + mbase + j < n_rows) op[(size_t)j * DOUT] = c[j];
    }
}

// ---------------------------------------------------------------------------
// log_softmax over rows of 64: one wave per row, 2 elements per lane
__global__ void log_softmax_kernel(const float* __restrict__ in,
                                   float* __restrict__ out, int n_rows) {
    int w    = (int)(((long long)blockIdx.x * blockDim.x + threadIdx.x) >> 5);
    int lane = threadIdx.x & 31;
    if (w >= n_rows) return;
    const float* row = in + (size_t)w * 64;
    float v0 = row[lane];
    float v1 = row[lane + 32];
    float m = fmaxf(v0, v1);
#pragma unroll
    for (int o = 16; o > 0; o >>= 1) m = fmaxf(m, __shfl_xor(m, o, 32));
    float s = expf(v0 - m) + expf(v1 - m);
#pragma unroll
    for (int o = 16; o > 0; o >>= 1) s += __shfl_xor(s, o, 32);
    float ls = logf(s);
    out[(size_t)w * 64 + lane]      = v0 - m - ls;
    out[(size_t)w * 64 + lane + 32] = v1 - m - ls;
}

// ---------------------------------------------------------------------------
extern "C" void kernel_launch(void* const* d_in, const int* in_sizes, int n_in,
                              void* d_out, int out_size, void* d_ws, size_t ws_size,
                              hipStream_t stream) {
    const float* x    = (const float*)d_in[0];
    const int*   ei   = (const int*)d_in[1];
    const float* W1l  = (const float*)d_in[2];
    const float* W1r  = (const float*)d_in[3];
    const float* b1   = (const float*)d_in[4];
    const float* Wml  = (const float*)d_in[5];
    const float* Wmr  = (const float*)d_in[6];
    const float* bm   = (const float*)d_in[7];
    const float* W2l  = (const float*)d_in[8];
    const float* W2r  = (const float*)d_in[9];
    const float* b2   = (const float*)d_in[10];

    const int N_ = in_sizes[0] / DIN;
    const int E_ = in_sizes[1] / 2;
    const int* src = ei;
    const int* dst = ei + E_;

    // workspace carve-out (256B aligned): deg | agg | h1 | h2 ; tmp aliases h1
    size_t off = 0;
    auto carve = [&](size_t bytes) {
        void* p = (char*)d_ws + off;
        off += (bytes + 255) & ~(size_t)255;
        return p;
    };
    float* deg = (float*)carve((size_t)N_ * 4);
    float* agg = (float*)carve((size_t)N_ * DIN * 4);
    float* h1  = (float*)carve((size_t)N_ * DIN * 4);
    float* h2  = (float*)carve((size_t)N_ * DIN * 4);
    float* tmp = h1;  // reuse: h1 dead after layer-2 GEMM

    const int rowBlocks  = (N_ + 15) / 16;
    const int scatBlocks = (E_ * 32 + 255) / 256;
    const int degBlocks  = (E_ + 255) / 256;
    const int lsmBlocks  = (N_ * 32 + 255) / 256;

    hipMemsetAsync(deg, 0, (size_t)N_ * 4, stream);
    degree_kernel<<<degBlocks, 256, 0, stream>>>(dst, deg, E_);

    // layer 1: x -> h1 (relu)
    hipMemsetAsync(agg, 0, (size_t)N_ * DIN * 4, stream);
    scatter_add_kernel<<<scatBlocks, 256, 0, stream>>>(x, src, dst, agg, E_);
    sage_gemm_kernel<8, 1><<<rowBlocks, 256, 0, stream>>>(agg, x, deg, W1l, W1r,
                                                          b1, h1, N_);

    // layer 2: h1 -> h2 (relu)
    hipMemsetAsync(agg, 0, (size_t)N_ * DIN * 4, stream);
    scatter_add_kernel<<<scatBlocks, 256, 0, stream>>>(h1, src, dst, agg, E_);
    sage_gemm_kernel<8, 1><<<rowBlocks, 256, 0, stream>>>(agg, h1, deg, Wml, Wmr,
                                                          bm, h2, N_);

    // layer 3: h2 -> tmp (64-wide, no relu), then log_softmax -> d_out
    hipMemsetAsync(agg, 0, (size_t)N_ * DIN * 4, stream);
    scatter_add_kernel<<<scatBlocks, 256, 0, stream>>>(h2, src, dst, agg, E_);
    sage_gemm_kernel<4, 0><<<rowBlocks, 128, 0, stream>>>(agg, h2, deg, W2l, W2r,
                                                          b2, tmp, N_);
    log_softmax_kernel<<<lsmBlocks, 256, 0, stream>>>(tmp, (float*)d_out, N_);
}